// Forces_43430709297374
// MI455X (gfx1250) — compile-verified
//
#include <hip/hip_runtime.h>

#define TILE   256
#define NCOPY  4
#define CPAD   (128*9 + 1)   /* 1153: pad so copies shift banks by 1 */

/* LDS staging layout per buffer, in dwords:
   dEd: 768 | rij: 768 | ii: 256 | jj: 256  => 2048 dwords = 8KB  */
#define BUF_DW  2048
#define OFF_DED 0
#define OFF_RIJ 768
#define OFF_II  1536
#define OFF_JJ  1792

/* gfx12-family CPol: TH in [2:0]; TH_NT = 1 (non-temporal).
   Edge streams are read exactly once (~320MB) - keep them out of L2 so the
   forces array (2.4MB, 60M atomics) and idx_m table (800KB) stay resident. */
#define CPOL_NT 1

#if defined(__HIP_DEVICE_COMPILE__) && defined(__gfx1250__) && \
    __has_builtin(__builtin_amdgcn_global_load_async_to_lds_b128) && \
    __has_builtin(__builtin_amdgcn_s_wait_asynccnt)
#define USE_ASYNC 1
#else
#define USE_ASYNC 0
#endif

#if USE_ASYNC
/* Builtin signature (from hipcc diagnostic):
   (v4i addrspace(1)*, v4i addrspace(3)*, imm i32 offset, imm i32 cpol) */
typedef int v4i __attribute__((vector_size(16)));
typedef __attribute__((address_space(1))) v4i gl_v4i;
typedef __attribute__((address_space(3))) v4i lds_v4i;
__device__ __forceinline__ gl_v4i*  to_gl(const void* p) { return (gl_v4i*)p;  }
__device__ __forceinline__ lds_v4i* to_lds(void* p)      { return (lds_v4i*)p; }
#endif

__device__ __forceinline__ void process_edge(
    float d0, float d1, float d2,
    float r0, float r1, float r2,
    int i, int j,
    const int* __restrict__ idx_m,
    float* __restrict__ forces,
    float* sstress, int myCopy)
{
    /* forces: F[i] += d ; F[j] -= d   (global f32 atomics, L2-resident) */
    atomicAdd(&forces[3*i+0],  d0);
    atomicAdd(&forces[3*i+1],  d1);
    atomicAdd(&forces[3*i+2],  d2);
    atomicAdd(&forces[3*j+0], -d0);
    atomicAdd(&forces[3*j+1], -d1);
    atomicAdd(&forces[3*j+2], -d2);
    /* stress[m][a][b] += r[a]*d[b]  into privatized LDS accumulator */
    int m = idx_m[i];
    float* s = sstress + myCopy + m*9;
    atomicAdd(&s[0], r0*d0); atomicAdd(&s[1], r0*d1); atomicAdd(&s[2], r0*d2);
    atomicAdd(&s[3], r1*d0); atomicAdd(&s[4], r1*d1); atomicAdd(&s[5], r1*d2);
    atomicAdd(&s[6], r2*d0); atomicAdd(&s[7], r2*d1); atomicAdd(&s[8], r2*d2);
}

#if USE_ASYNC
/* Issue async global->LDS loads for one full tile of 256 edges.
   512 x 16B chunks total; each thread issues exactly 2 b128 loads.
   Branch boundary at t==192 is wave-aligned (192 = 6*32): no divergence,
   every wave has exactly 2 outstanding ASYNCcnt entries per tile. */
__device__ __forceinline__ void issue_tile(
    const float* __restrict__ dEd, const float* __restrict__ rij,
    const int* __restrict__ ii,   const int* __restrict__ jj,
    float* buf, int tile, int t)
{
    const int base = tile * TILE;
    if (t < 192) {
        __builtin_amdgcn_global_load_async_to_lds_b128(
            to_gl(dEd + (size_t)base*3 + t*4),
            to_lds(buf + OFF_DED + t*4), 0, CPOL_NT);
        __builtin_amdgcn_global_load_async_to_lds_b128(
            to_gl(rij + (size_t)base*3 + t*4),
            to_lds(buf + OFF_RIJ + t*4), 0, CPOL_NT);
    } else {
        const int c = t - 192;
        __builtin_amdgcn_global_load_async_to_lds_b128(
            to_gl(ii + (size_t)base + c*4),
            to_lds(buf + OFF_II + c*4), 0, CPOL_NT);
        __builtin_amdgcn_global_load_async_to_lds_b128(
            to_gl(jj + (size_t)base + c*4),
            to_lds(buf + OFF_JJ + c*4), 0, CPOL_NT);
    }
}
#endif

__global__ __launch_bounds__(256)
void edge_scatter_kernel(const float* __restrict__ dEd,
                         const float* __restrict__ rij,
                         const int*   __restrict__ idx_i,
                         const int*   __restrict__ idx_j,
                         const int*   __restrict__ idx_m,
                         float* __restrict__ forces,
                         float* __restrict__ wstress,
                         int E)
{
    __shared__ float sstress[NCOPY * CPAD];
#if USE_ASYNC
    __shared__ float sbuf[2 * BUF_DW];
#endif
    const int t = threadIdx.x;
    for (int s = t; s < NCOPY * CPAD; s += 256) sstress[s] = 0.0f;
    __syncthreads();

    const int myCopy = (t & (NCOPY - 1)) * CPAD;
    const int tileCount = (E + TILE - 1) / TILE;

#if USE_ASYNC
    int tile = blockIdx.x;
    int pb = 0;
    if (tile < tileCount && (tile * TILE + TILE) <= E)
        issue_tile(dEd, rij, idx_i, idx_j, sbuf, tile, t);

    for (; tile < tileCount; tile += gridDim.x) {
        const int base = tile * TILE;
        const bool full = (base + TILE) <= E;
        if (full) {
            const int nextTile = tile + gridDim.x;
            const bool nextFull =
                (nextTile < tileCount) && ((nextTile * TILE + TILE) <= E);
            if (nextFull) {
                issue_tile(dEd, rij, idx_i, idx_j,
                           sbuf + (pb ^ 1) * BUF_DW, nextTile, t);
                __builtin_amdgcn_s_wait_asynccnt(2);  /* current tile done */
            } else {
                __builtin_amdgcn_s_wait_asynccnt(0);
            }
            __syncthreads();                      /* all waves' data in LDS */
            const float* db = sbuf + pb * BUF_DW;
            const int*   ib = (const int*)db;
            float d0 = db[OFF_DED + t*3 + 0];
            float d1 = db[OFF_DED + t*3 + 1];
            float d2 = db[OFF_DED + t*3 + 2];
            float r0 = db[OFF_RIJ + t*3 + 0];
            float r1 = db[OFF_RIJ + t*3 + 1];
            float r2 = db[OFF_RIJ + t*3 + 2];
            int   i  = ib[OFF_II + t];
            int   j  = ib[OFF_JJ + t];
            process_edge(d0,d1,d2, r0,r1,r2, i, j, idx_m, forces, sstress, myCopy);
            __syncthreads();                      /* safe to overwrite buf */
            pb ^= 1;
        } else {
            /* global tail tile: direct nontemporal loads */
            const int e = base + t;
            if (e < E) {
                float d0 = __builtin_nontemporal_load(&dEd[(size_t)e*3 + 0]);
                float d1 = __builtin_nontemporal_load(&dEd[(size_t)e*3 + 1]);
                float d2 = __builtin_nontemporal_load(&dEd[(size_t)e*3 + 2]);
                float r0 = __builtin_nontemporal_load(&rij[(size_t)e*3 + 0]);
                float r1 = __builtin_nontemporal_load(&rij[(size_t)e*3 + 1]);
                float r2 = __builtin_nontemporal_load(&rij[(size_t)e*3 + 2]);
                int   i  = __builtin_nontemporal_load(&idx_i[e]);
                int   j  = __builtin_nontemporal_load(&idx_j[e]);
                process_edge(d0,d1,d2, r0,r1,r2, i, j, idx_m, forces, sstress, myCopy);
            }
        }
    }
#else
    for (int e = blockIdx.x * 256 + t; e < E; e += gridDim.x * 256) {
        float d0 = __builtin_nontemporal_load(&dEd[(size_t)e*3 + 0]);
        float d1 = __builtin_nontemporal_load(&dEd[(size_t)e*3 + 1]);
        float d2 = __builtin_nontemporal_load(&dEd[(size_t)e*3 + 2]);
        float r0 = __builtin_nontemporal_load(&rij[(size_t)e*3 + 0]);
        float r1 = __builtin_nontemporal_load(&rij[(size_t)e*3 + 1]);
        float r2 = __builtin_nontemporal_load(&rij[(size_t)e*3 + 2]);
        int   i  = __builtin_nontemporal_load(&idx_i[e]);
        int   j  = __builtin_nontemporal_load(&idx_j[e]);
        process_edge(d0,d1,d2, r0,r1,r2, i, j, idx_m, forces, sstress, myCopy);
    }
    (void)tileCount;
#endif

    /* flush block-local stress accumulator (sum over NCOPY copies) */
    __syncthreads();
    for (int s = t; s < 128*9; s += 256) {
        float v = 0.0f;
        #pragma unroll
        for (int c = 0; c < NCOPY; ++c) v += sstress[c * CPAD + s];
        atomicAdd(&wstress[s], v);
    }
}

__global__ __launch_bounds__(256)
void zero_kernel(float* __restrict__ out, int nOut,
                 float* __restrict__ ws, int nWs)
{
    const int i = blockIdx.x * blockDim.x + threadIdx.x;
    const int stride = gridDim.x * blockDim.x;
    for (int k = i; k < nOut; k += stride) out[k] = 0.0f;
    for (int k = i; k < nWs;  k += stride) ws[k]  = 0.0f;
}

__global__ __launch_bounds__(128)
void finalize_kernel(const float* __restrict__ cell,
                     const float* __restrict__ wstress,
                     float* __restrict__ stress_out, int M)
{
    const int m = threadIdx.x;
    if (m < M) {
        const float* c = cell + (size_t)m * 9;
        const float a0=c[0], a1=c[1], a2=c[2];
        const float b0=c[3], b1=c[4], b2=c[5];
        const float e0=c[6], e1=c[7], e2=c[8];
        const float cx = b1*e2 - b2*e1;
        const float cy = b2*e0 - b0*e2;
        const float cz = b0*e1 - b1*e0;
        const float inv = 1.0f / (a0*cx + a1*cy + a2*cz);
        #pragma unroll
        for (int k = 0; k < 9; ++k)
            stress_out[(size_t)m*9 + k] = wstress[(size_t)m*9 + k] * inv;
    }
}

extern "C" void kernel_launch(void* const* d_in, const int* in_sizes, int n_in,
                              void* d_out, int out_size, void* d_ws, size_t ws_size,
                              hipStream_t stream)
{
    const float* dEd   = (const float*)d_in[0];
    const float* rij   = (const float*)d_in[1];
    /* d_in[2] = R (unused by the math) */
    const float* cell  = (const float*)d_in[3];
    const int*   idx_i = (const int*)d_in[4];
    const int*   idx_j = (const int*)d_in[5];
    const int*   idx_m = (const int*)d_in[6];

    const int E = in_sizes[0] / 3;
    const int N = in_sizes[2] / 3;
    const int M = in_sizes[3] / 9;

    float* forces     = (float*)d_out;
    float* stress_out = (float*)d_out + (size_t)N * 3;
    float* ws         = (float*)d_ws;   /* M*9 accumulator, zeroed each call */

    zero_kernel<<<512, 256, 0, stream>>>(forces, N * 3, ws, M * 9);

    const int tileCount = (E + TILE - 1) / TILE;
    int grid = 2048;
    if (grid > tileCount) grid = tileCount;
    if (grid < 1) grid = 1;
    edge_scatter_kernel<<<grid, 256, 0, stream>>>(
        dEd, rij, idx_i, idx_j, idx_m, forces, ws, E);

    finalize_kernel<<<1, 128, 0, stream>>>(cell, ws, stress_out, M);
}